// DarkNet53_45397804319218
// MI455X (gfx1250) — compile-verified
//
#include <hip/hip_runtime.h>
#include <hip/hip_bf16.h>

// ---------------- problem constants (from reference) ----------------
#define Bn      32
#define An      3
#define NCLS    80
#define ATTRN   85            // 5 + NCLS
#define GH      76
#define GW      76
#define SPW     (GH * GW)     // 5776 spatial positions per (b, a, attr) plane
#define WPG     8.0f          // 608 / 76
#define HPG     8.0f

#define TPOS    GW            // tile = one grid row: 76 spatial positions
#define TELEM   (TPOS * ATTRN) // 6460 floats per tile
#define TV4     (TELEM / 4)    // 1615 float4 (exact)
#define THREADS 256

// typed address-space pointers for the gfx1250 async-LDS builtins
// (builtin signature wants AS(1) int* / AS(3) int*, per hipcc diagnostic)
typedef __attribute__((address_space(1))) int g_int;
typedef __attribute__((address_space(3))) int l_int;

#if defined(__has_builtin)
#  if __has_builtin(__builtin_amdgcn_global_load_async_to_lds_b32)
#    define HAVE_ASYNC_LDS 1
#  else
#    define HAVE_ASYNC_LDS 0
#  endif
#else
#  define HAVE_ASYNC_LDS 0
#endif

__device__ __forceinline__ float sig_(float x) {
    return 1.0f / (1.0f + __expf(-x));
}

__global__ __launch_bounds__(THREADS)
void yolo_decode_kernel(const float* __restrict__ feat,
                        const float* __restrict__ anchors,
                        float* __restrict__ out)
{
    __shared__ __align__(16) float tile[TELEM];

    const int bid = blockIdx.x;            // over Bn * An * GH
    const int j   = bid % GH;
    const int a   = (bid / GH) % An;
    const int b   = bid / (GH * An);
    const int tid = threadIdx.x;

    const float aw = anchors[2 * a + 0];
    const float ah = anchors[2 * a + 1];

    // input plane base for (b, a): [ATTRN][SPW], we take row j of each attr
    const float* src0 = feat
        + ((size_t)b * (An * ATTRN) + (size_t)a * ATTRN) * SPW
        + (size_t)j * GW;

    // ---- Phase 1: stage raw tile into LDS, transposed on the fly ----
    // global [c][i] (contiguous in i)  ->  LDS [i][c] (stride 85, coprime
    // with 64 banks => conflict-free scatter)
#if HAVE_ASYNC_LDS
    for (int t = tid; t < TELEM; t += THREADS) {
        const int c = t / TPOS;
        const int i = t - c * TPOS;
        const float* g = src0 + (size_t)c * SPW + i;
        float*       l = &tile[i * ATTRN + c];
        __builtin_amdgcn_global_load_async_to_lds_b32(
            (g_int*)g, (l_int*)l, /*offset=*/0, /*cpol=*/0);
    }
#  if __has_builtin(__builtin_amdgcn_s_wait_asynccnt)
    __builtin_amdgcn_s_wait_asynccnt(0);
#  else
    asm volatile("s_wait_asynccnt 0" ::: "memory");
#  endif
    __syncthreads();
#else
    for (int t = tid; t < TELEM; t += THREADS) {
        const int c = t / TPOS;
        const int i = t - c * TPOS;
        tile[i * ATTRN + c] = src0[(size_t)c * SPW + i];
    }
    __syncthreads();
#endif

    // ---- Phase 2: decode attrs 0..4, stream tile out contiguously ----
    // out row for (b, a, j, i) starts at ((b*An + a)*SPW + j*GW + i)*ATTRN
    const size_t outBase =
        ((size_t)b * (An * SPW) + (size_t)a * SPW + (size_t)j * GW) * ATTRN;
    float4*       out4 = (float4*)(out + outBase);   // 16B-aligned (proven)
    const float4* lds4 = (const float4*)tile;

    for (int q = tid; q < TV4; q += THREADS) {
        float4 v = lds4[q];                          // linear: conflict-free
        float* vp = (float*)&v;
        const int e0 = q * 4;
#pragma unroll
        for (int k = 0; k < 4; ++k) {
            const int e = e0 + k;
            const int i = e / ATTRN;                 // column index in grid
            const int c = e - i * ATTRN;             // attribute index
            float x = vp[k];
            if (c == 0)      x = sig_(x);                          // objness
            else if (c == 1) x = (sig_(x) + (float)i) * WPG;       // box_x
            else if (c == 2) x = (sig_(x) + (float)j) * HPG;       // box_y
            else if (c == 3) x = aw * __expf(x) * WPG;             // box_w
            else if (c == 4) x = ah * __expf(x) * HPG;             // box_h
            vp[k] = x;                               // c >= 5: raw class conf
        }
        out4[q] = v;                                 // contiguous b128 store
    }
}

extern "C" void kernel_launch(void* const* d_in, const int* in_sizes, int n_in,
                              void* d_out, int out_size, void* d_ws, size_t ws_size,
                              hipStream_t stream)
{
    const float* feat    = (const float*)d_in[0];   // [Bn, An*ATTRN, GH, GW]
    const float* anchors = (const float*)d_in[1];   // [2*An]
    float*       out     = (float*)d_out;           // [Bn, An*GH*GW, ATTRN]

    const int nblocks = Bn * An * GH;               // 7296 workgroups
    yolo_decode_kernel<<<dim3(nblocks), dim3(THREADS), 0, stream>>>(
        feat, anchors, out);
}